// TransformerDecoderKVCache_26972394619628
// MI455X (gfx1250) — compile-verified
//
#include <hip/hip_runtime.h>
#include <stdint.h>

typedef uint32_t u32;
typedef uint64_t u64;
typedef __attribute__((ext_vector_type(4))) u32 v4u;
typedef __attribute__((ext_vector_type(4))) int v4i;
typedef __attribute__((ext_vector_type(8))) int v8i;

// Problem shape (fixed by the reference harness).
constexpr u32 B      = 4;
constexpr u32 S_PAST = 4096;
constexpr u32 S_NEW  = 16;
constexpr u32 H      = 32;
constexpr u32 D      = 128;
constexpr u32 HD     = H * D;              // 4096 floats per (b, s) row
constexpr u32 S_OUT  = S_PAST + S_NEW;     // 4112

// TDM chunking: 2048 fp32 = 8 KB per DMA op. Every region length and every
// src/dst chunk offset is a multiple of 2048 floats (8 KB aligned).
constexpr u32 CH           = 2048;
constexpr u32 CACHE_CHUNKS = B * S_PAST * HD / CH;   // 32768 per tensor
constexpr u32 NEW_CHUNKS   = B * S_NEW  * HD / CH;   // 128 per tensor
constexpr u32 PER_TENSOR   = CACHE_CHUNKS + NEW_CHUNKS;
constexpr u32 TOTAL_CHUNKS = 2 * PER_TENSOR;         // 65792 (K and V)

constexpr u32 WAVES_PER_BLOCK = 4;
constexpr u32 BLOCK = 32 * WAVES_PER_BLOCK;          // wave32!
constexpr u32 GRID  = 1024;                          // 4096 waves total

// ---------------------------------------------------------------------------
// Tensor Data Mover: one DMA op moves an 8 KB 1-D tile between global and LDS.
// D# layout per CDNA5 ISA.
// ---------------------------------------------------------------------------
__device__ __forceinline__ void tdm_op(bool is_store, u32 lds_off, u64 gaddr) {
  v4u g0;
  g0.x = 1u;                                   // count=1 valid descriptor
  g0.y = lds_off;                              // LDS byte address of tile
  g0.z = (u32)(gaddr & 0xFFFFFFFFu);           // global_addr[31:0]
  g0.w = (u32)((gaddr >> 32) & 0x1FFFFFFu)     // global_addr[56:32]
       | (2u << 30);                           // type=2 ("image")

  v8i g1;
  g1[0] = (int)(2u << 16);                     // workgroup_mask=0, data_size=2 (4B)
  g1[1] = (int)((CH & 0xFFFFu) << 16);         // tensor_dim0[15:0]   @ [63:48]
  g1[2] = (int)(((CH >> 16) & 0xFFFFu)         // tensor_dim0[31:16]  @ [79:64]
       | (1u << 16));                          // tensor_dim1[15:0]=1 @ [95:80]
  g1[3] = (int)((CH & 0xFFFFu) << 16);         // tensor_dim1[31:16]=0; tile_dim0 @ [127:112]
  g1[4] = 0;                                   // tile_dim1=0, tile_dim2=0 (1-D tile)
  g1[5] = (int)CH;                             // tensor_dim0_stride[31:0]
  g1[6] = 0;                                   // stride0[47:32]=0, stride1[15:0]=0
  g1[7] = 0;                                   // stride1[47:16]=0

  // Benign groups 2/3: higher dims = 1 so OOB checks can never fire.
  v4i g2; g2[0] = 1; g2[1] = 1; g2[2] = 0; g2[3] = 0;   // tensor_dim2=1, tensor_dim3=1
  v4i g3; g3[0] = 0; g3[1] = (int)(1u << 16); g3[2] = 0; g3[3] = 0; // tensor_dim4=1

#if __clang_major__ >= 23
  v8i gx; gx[0]=0; gx[1]=0; gx[2]=0; gx[3]=0; gx[4]=0; gx[5]=0; gx[6]=0; gx[7]=0;
  if (is_store) __builtin_amdgcn_tensor_store_from_lds(g0, g1, g2, g3, gx, 0);
  else          __builtin_amdgcn_tensor_load_to_lds  (g0, g1, g2, g3, gx, 0);
#else
  if (is_store) __builtin_amdgcn_tensor_store_from_lds(g0, g1, g2, g3, 0);
  else          __builtin_amdgcn_tensor_load_to_lds  (g0, g1, g2, g3, 0);
#endif
}

// Map a global chunk index to (src pointer, dst pointer).
__device__ __forceinline__ void chunk_addrs(u32 c,
                                            const float* kc, const float* vc,
                                            const float* kn, const float* vn,
                                            float* out,
                                            const float** src, float** dst) {
  const float* cache_base;
  const float* new_base;
  float* obase;
  if (c < PER_TENSOR) {                  // K tensor
    cache_base = kc; new_base = kn; obase = out;
  } else {                               // V tensor
    c -= PER_TENSOR;
    cache_base = vc; new_base = vn;
    obase = out + (u64)B * S_OUT * HD;
  }
  if (c < CACHE_CHUNKS) {
    constexpr u32 cpb = CACHE_CHUNKS / B;          // 8192 chunks per batch
    u32 b = c / cpb, i = c % cpb;
    *src = cache_base + (u64)c * CH;               // cache is contiguous
    *dst = obase + (u64)b * S_OUT * HD + (u64)i * CH;
  } else {
    u32 t = c - CACHE_CHUNKS;
    constexpr u32 cpb = NEW_CHUNKS / B;            // 32 chunks per batch
    u32 b = t / cpb, i = t % cpb;
    *src = new_base + (u64)t * CH;
    *dst = obase + (u64)b * S_OUT * HD + (u64)S_PAST * HD + (u64)i * CH;
  }
}

__global__ __launch_bounds__(BLOCK) void kv_append_tdm(
    const float* __restrict__ kc, const float* __restrict__ vc,
    const float* __restrict__ kn, const float* __restrict__ vn,
    float* __restrict__ out) {
  // Per-wave double buffer: 2 x 8 KB. 4 waves -> 64 KB static LDS per block.
  __shared__ __align__(1024) float lds[WAVES_PER_BLOCK][2][CH];

  const u32 wave_in_blk = threadIdx.x >> 5;
  const u32 wave  = blockIdx.x * WAVES_PER_BLOCK + wave_in_blk;
  const u32 nwaves = gridDim.x * WAVES_PER_BLOCK;

  // Flat LDS pointers: low 32 bits are the workgroup-relative LDS byte address.
  const u32 lds_buf[2] = {
    (u32)(uintptr_t)&lds[wave_in_blk][0][0],
    (u32)(uintptr_t)&lds[wave_in_blk][1][0]
  };

  u32 c = wave;
  if (c >= TOTAL_CHUNKS) return;   // wave-uniform exit

  const float* src; float* dst;
  chunk_addrs(c, kc, vc, kn, vn, out, &src, &dst);
  tdm_op(false, lds_buf[0], (u64)(uintptr_t)src);      // prime buffer 0

  u32    pend_lds = lds_buf[0];
  float* pend_dst = dst;
  u32    buf = 1;

  for (c += nwaves; c < TOTAL_CHUNKS; c += nwaves) {
    chunk_addrs(c, kc, vc, kn, vn, out, &src, &dst);
    // Wait for: load into pend_lds (we store from it) and the store that
    // previously read lds_buf[buf] (we overwrite it). Tensor ops from one
    // wave complete in order, so tensorcnt==0 covers both.
    __builtin_amdgcn_s_wait_tensorcnt(0);
    tdm_op(true,  pend_lds,     (u64)(uintptr_t)pend_dst);  // LDS -> global
    tdm_op(false, lds_buf[buf], (u64)(uintptr_t)src);       // global -> LDS
    pend_lds = lds_buf[buf];
    pend_dst = dst;
    buf ^= 1u;
  }

  __builtin_amdgcn_s_wait_tensorcnt(0);
  tdm_op(true, pend_lds, (u64)(uintptr_t)pend_dst);
  // s_endpgm performs an implicit wait-idle, covering the final store.
}

extern "C" void kernel_launch(void* const* d_in, const int* in_sizes, int n_in,
                              void* d_out, int out_size, void* d_ws, size_t ws_size,
                              hipStream_t stream) {
  const float* kc = (const float*)d_in[0];  // k_cache [B, S_PAST, H, D]
  const float* vc = (const float*)d_in[1];  // v_cache [B, S_PAST, H, D]
  const float* kn = (const float*)d_in[2];  // k       [B, S_NEW,  H, D]
  const float* vn = (const float*)d_in[3];  // v       [B, S_NEW,  H, D]
  float* out = (float*)d_out;               // [k_out ; v_out] concatenated

  kv_append_tdm<<<dim3(GRID), dim3(BLOCK), 0, stream>>>(kc, vc, kn, vn, out);
}